// KCL_56195352101242
// MI455X (gfx1250) — compile-verified
//
#include <hip/hip_runtime.h>
#include <stdint.h>

#define QLEN   65536
#define NTOT   65537        // Q + 1
#define TPB    256
#define NWAVE  8
#define NUM_POS 6
#define INV_T  (1.0f / 0.07f)
#define CHUNK  (QLEN / TPB) // 256 contiguous queue columns per thread

typedef __attribute__((ext_vector_type(2))) float v2f;
typedef __attribute__((ext_vector_type(8))) float v8f;

// ---------------- JAX threefry2x32 (exact) ----------------
__device__ __forceinline__ void threefry2x32(uint32_t k0, uint32_t k1,
                                             uint32_t x0, uint32_t x1,
                                             uint32_t& o0, uint32_t& o1) {
  uint32_t ks[3] = {k0, k1, k0 ^ k1 ^ 0x1BD11BDAu};
  const uint32_t rotA[4] = {13u, 15u, 26u, 6u};
  const uint32_t rotB[4] = {17u, 29u, 16u, 24u};
  x0 += ks[0]; x1 += ks[1];
#pragma unroll
  for (int i = 0; i < 5; ++i) {
    const uint32_t* r = (i & 1) ? rotB : rotA;
#pragma unroll
    for (int j = 0; j < 4; ++j) {
      x0 += x1;
      x1 = (x1 << r[j]) | (x1 >> (32u - r[j]));
      x1 ^= x0;
    }
    x0 += ks[(i + 1) % 3];
    x1 += ks[(i + 2) % 3] + (uint32_t)(i + 1);
  }
  o0 = x0; o1 = x1;
}

// element b of jax.random.uniform(key, (256,), float32)
__device__ __forceinline__ float jax_uniform256(uint32_t k0, uint32_t k1, int b) {
  uint32_t o0, o1, bits;
  if (b < 128) { threefry2x32(k0, k1, (uint32_t)b, (uint32_t)(b + 128), o0, o1); bits = o0; }
  else         { threefry2x32(k0, k1, (uint32_t)(b - 128), (uint32_t)b, o0, o1); bits = o1; }
  bits = (bits >> 9) | 0x3F800000u;
  return __uint_as_float(bits) - 1.0f;
}

// ---------------- per-row kernel: one workgroup per logits row ----------------
extern "C" __global__ __launch_bounds__(TPB)
void moco_loss_rows(const float* __restrict__ logits,
                    const int*   __restrict__ im_labels,
                    const int*   __restrict__ queue_label,
                    float*       __restrict__ row_loss) {
  extern __shared__ char lds_dyn[];          // holds the (shifted) row: ~256 KB
  __shared__ float s_red[16];                // wave partials
  __shared__ float s_bcast[4];               // [0]=row max, [1]=lse
  __shared__ int   s_wsum[8];                // per-wave positive counts
  __shared__ int   s_woff[8];                // per-wave exclusive offsets
  __shared__ int   s_np[2];                  // [0]=n_pos
  __shared__ int   s_targ[8];
  __shared__ int   s_col[8];

  const int b    = blockIdx.x;
  const int tid  = threadIdx.x;
  const int lane = tid & 31;
  const int wave = tid >> 5;

  // ---- Phase A: issue async DMA of the whole row into LDS (16B-aligned) ----
  const uint64_t gbase   = (uint64_t)(uintptr_t)(logits + (size_t)b * NTOT);
  const unsigned lds_raw = (unsigned)(uintptr_t)lds_dyn;   // low 32 bits = LDS byte offset
  const unsigned lds16   = (lds_raw + 15u) & ~15u;
  const unsigned pad     = lds16 - lds_raw;
  const unsigned shift   = (unsigned)(gbase & 15u);        // rows stride 4 mod 16
  const uint64_t gal     = gbase & ~(uint64_t)15;
  const unsigned chunks  = (shift + (unsigned)NTOT * 4u + 15u) >> 4;  // <= 16385

  const float* rowl = (const float*)(lds_dyn + pad + shift);   // element i of the row

  for (unsigned it = tid; it < chunks; it += TPB) {
    unsigned off   = it * 16u;
    unsigned laddr = lds16 + off;
    asm volatile("global_load_async_to_lds_b128 %0, %1, %2 th:TH_LOAD_NT"
                 :: "v"(laddr), "v"(off), "s"(gal) : "memory");
  }
  if (tid < NUM_POS) s_col[tid] = -1;

  // ================= overlapped with the DMA: label/sampling work =================

  // ---- Phase B: positives count per contiguous 256-column chunk ----
  const int label = im_labels[b];
  const int c0 = tid * CHUNK;
  int cnt = 0;
  for (int j = 0; j < CHUNK; ++j) cnt += (queue_label[c0 + j] == label) ? 1 : 0;

  // wave-level inclusive scan of counts
  int inc = cnt;
#pragma unroll
  for (int d = 1; d < 32; d <<= 1) {
    int t = __shfl_up(inc, d);
    if (lane >= d) inc += t;
  }
  if (lane == 31) s_wsum[wave] = inc;
  __syncthreads();
  if (tid == 0) {
    int run = 0;
    for (int w = 0; w < NWAVE; ++w) { s_woff[w] = run; run += s_wsum[w]; }
    s_np[0] = run;
  }
  __syncthreads();
  const int myPre = s_woff[wave] + inc - cnt;   // exclusive prefix of this thread

  // ---- Phase C: 6 threefry samples in parallel (threads 0..5) ----
  if (tid < NUM_POS) {
    uint32_t fk0, fk1;
    threefry2x32(0u, 1u, 0u, (uint32_t)tid, fk0, fk1);  // fold_in(key(1), i)
    const float r = jax_uniform256(fk0, fk1, b);
    const int np = s_np[0];
    int ji = (np > 0) ? (int)floorf(r * (float)np) : -1;
    if (ji >= np) ji = -1;
    s_targ[tid] = ji;
  }
  __syncthreads();

  // ---- Phase D: locate the j-th positive's column (owner thread rescans) ----
  for (int i = 0; i < NUM_POS; ++i) {
    const int tj = s_targ[i];
    if (tj >= myPre && tj < myPre + cnt) {
      int need = tj - myPre;
      int seen = 0;
      for (int j = 0; j < CHUNK; ++j) {
        if (queue_label[c0 + j] == label) {
          if (seen == need) { s_col[i] = c0 + j; break; }
          ++seen;
        }
      }
    }
  }

  // ================= row data now needed: drain the DMA =================
  asm volatile("s_wait_asynccnt 0" ::: "memory");
  __syncthreads();

  // ---- Phase E: row max (raw logits; scale commutes with max) ----
  float m = -3.402823466e38f;
  for (int i = tid; i < NTOT; i += TPB) m = fmaxf(m, rowl[i]);
#pragma unroll
  for (int d = 16; d >= 1; d >>= 1) m = fmaxf(m, __shfl_xor(m, d));
  if (lane == 0) s_red[wave] = m;
  __syncthreads();
  if (tid == 0) {
    float mm = s_red[0];
    for (int w = 1; w < NWAVE; ++w) mm = fmaxf(mm, s_red[w]);
    s_bcast[0] = mm;
  }
  __syncthreads();
  const float Mraw = s_bcast[0];

  // ---- Phase F: sum of exp via WMMA f32 16x16x4 with B = ones ----
  // 1024 tiles x 64 elements cover the first 65536 elements; last element scalar.
  // Arg is <= 0, so the fast __expf (single v_exp_f32) is safe.
  v8f acc = {0.f, 0.f, 0.f, 0.f, 0.f, 0.f, 0.f, 0.f};
  v2f bones; bones.x = 1.0f; bones.y = 1.0f;
  for (int t = wave; t < 1024; t += NWAVE) {
    const int base = t * 64;
    v2f a;
    a.x = __expf((rowl[base + lane]      - Mraw) * INV_T);
    a.y = __expf((rowl[base + 32 + lane] - Mraw) * INV_T);
    acc = __builtin_amdgcn_wmma_f32_16x16x4_f32(false, a, false, bones,
                                                (short)0, acc, false, false);
  }
  // D[m,n] = rowsum_m for every n  =>  sum(all 256 entries) = 16 * wave partial
  float ps = acc[0] + acc[1] + acc[2] + acc[3] + acc[4] + acc[5] + acc[6] + acc[7];
#pragma unroll
  for (int d = 16; d >= 1; d >>= 1) ps += __shfl_xor(ps, d);
  if (lane == 0) s_red[wave] = ps * (1.0f / 16.0f);
  __syncthreads();
  if (tid == 0) {
    float S = __expf((rowl[QLEN] - Mraw) * INV_T);   // tail element 65536
    for (int w = 0; w < NWAVE; ++w) S += s_red[w];
    s_bcast[1] = Mraw * INV_T + logf(S);             // logsumexp of scaled logits
  }
  __syncthreads();

  // ---- Phase G: per-row loss ----
  if (tid == 0) {
    const float lse = s_bcast[1];
    float inner = rowl[0] * INV_T - lse;   // the "ones" column (col 0)
    int msum = 1;
    for (int i = 0; i < NUM_POS; ++i) {
      const int c = s_col[i];
      if (c < 0) continue;
      bool dup = false;
      for (int k2 = 0; k2 < i; ++k2) if (s_col[k2] == c) { dup = true; break; }
      if (!dup) { inner += rowl[c + 1] * INV_T - lse; ++msum; }
    }
    row_loss[b] = -(inner / (float)msum) * (1.0f / (float)TPB);
  }
}

// ---------------- final deterministic reduction of 256 row losses ----------------
extern "C" __global__ __launch_bounds__(TPB)
void moco_loss_reduce(const float* __restrict__ row_loss, float* __restrict__ out) {
  __shared__ float sb[8];
  const int tid = threadIdx.x;
  float v = row_loss[tid];
#pragma unroll
  for (int d = 16; d >= 1; d >>= 1) v += __shfl_xor(v, d);
  if ((tid & 31) == 0) sb[tid >> 5] = v;
  __syncthreads();
  if (tid == 0) {
    float s = 0.f;
    for (int w = 0; w < 8; ++w) s += sb[w];
    *out = s;
  }
}

extern "C" void kernel_launch(void* const* d_in, const int* in_sizes, int n_in,
                              void* d_out, int out_size, void* d_ws, size_t ws_size,
                              hipStream_t stream) {
  const float* logits      = (const float*)d_in[0];
  const int*   im_labels   = (const int*)d_in[1];
  const int*   queue_label = (const int*)d_in[2];
  float* out = (float*)d_out;
  float* ws  = (float*)d_ws;   // 256 floats of per-row loss

  // dynamic LDS: 16 (align pad) + 12 (shift) + 16385 x 16B chunks
  const size_t shbytes = (size_t)NTOT * 4 + 64;   // 262212 B < 320 KB WGP LDS

  moco_loss_rows<<<256, TPB, shbytes, stream>>>(logits, im_labels, queue_label, ws);
  moco_loss_reduce<<<1, TPB, 0, stream>>>(ws, out);
}